// Sn_coarsen_layer_6296422056040
// MI455X (gfx1250) — compile-verified
//
#include <hip/hip_runtime.h>

typedef __attribute__((ext_vector_type(2))) float v2f;
typedef __attribute__((ext_vector_type(8))) float v8f;

#define C_    16
#define N_    256
#define IN_   32
#define OUT_  32
#define BS_   64
#define K_    512   // IN_ * C_   (contraction dim: (i, c_q))
#define NP_   512   // C_ * OUT_  (output dim: (c_p, o))

// ---------------------------------------------------------------------------
// Step A: cluster-sum of permuted input.
// A[b, i*16 + c] = sum_{s<16} x[b, c + 16*s, i]
// x flat layout: [BS, N, IN] -> b*8192 + n*32 + i
// ---------------------------------------------------------------------------
__global__ void build_A(const float* __restrict__ x, float* __restrict__ A) {
    int t = blockIdx.x * blockDim.x + threadIdx.x;   // 0 .. BS_*K_
    if (t >= BS_ * K_) return;
    int b = t >> 9;          // /512
    int k = t & 511;
    int i = k >> 4;
    int c = k & 15;
    const float* xp = x + (size_t)b * (N_ * IN_) + c * IN_ + i;
    float s = 0.0f;
#pragma unroll
    for (int ss = 0; ss < 16; ++ss) s += xp[ss * (C_ * IN_)];   // stride 512 floats
    A[t] = s;
}

// ---------------------------------------------------------------------------
// Step B: fused weight matrix.
// Bmat[k=(i,c_q), n=(c_p,o)] = w_off[o*32+i, c_p, c_q] / 256
//                             + (c_q==c_p) * w_diag[o*32+i, c_p]
// ---------------------------------------------------------------------------
__global__ void build_B(const float* __restrict__ w_diag,
                        const float* __restrict__ w_off,
                        float* __restrict__ B) {
    int t = blockIdx.x * blockDim.x + threadIdx.x;   // 0 .. K_*NP_
    if (t >= K_ * NP_) return;
    int k = t >> 9;
    int n = t & 511;
    int i  = k >> 4, cq = k & 15;
    int cp = n >> 5, o  = n & 31;
    int wrow = o * IN_ + i;                           // 0..1023
    float v = w_off[(size_t)wrow * 256 + cp * 16 + cq] * (1.0f / 256.0f);
    if (cq == cp) v += w_diag[wrow * 16 + cp];
    B[t] = v;
}

// ---------------------------------------------------------------------------
// Step C: 64x512x512 fp32 GEMM via V_WMMA_F32_16X16X4_F32, one wave per
// 16x16 tile of out_cluster; epilogue adds bias and broadcasts each cluster
// value to its 16 original positions (n = c_p + 16*s).
//
// A-frag layout (ISA 7.12.2, 32-bit A 16x4): lanes 0-15 hold M=lane with
// VGPR0=K0,VGPR1=K1; lanes 16-31 hold M=lane-16 with VGPR0=K2,VGPR1=K3.
// B-frag mirrors this per column. C/D: VGPR r -> M = r + 8*(lane>=16),
// N = lane&15.
// ---------------------------------------------------------------------------
__global__ void gemm_wmma(const float* __restrict__ A,
                          const float* __restrict__ B,
                          const float* __restrict__ b1,
                          float* __restrict__ out) {
    const int tn   = blockIdx.x;        // 0..31  (n'-tile)
    const int tm   = blockIdx.y;        // 0..3   (batch tile)
    const int lane = threadIdx.x;       // 0..31
    const int half = lane >> 4;         // 0 or 1
    const int l    = lane & 15;

    // per-lane base pointers (fold the half*2 K-offset in)
    const float* Arow = A + (size_t)(tm * 16 + l) * K_ + half * 2;
    const float* Bcol = B + (tn * 16 + l) + (size_t)(half * 2) * NP_;

    v8f acc = {};
#pragma unroll 4
    for (int k0 = 0; k0 < K_; k0 += 4) {
        v2f a, b;
        // contiguous pair -> single global_load_b64
        const float2 av = *reinterpret_cast<const float2*>(Arow + k0);
        a.x = av.x; a.y = av.y;
        b.x = Bcol[(size_t)k0 * NP_];
        b.y = Bcol[(size_t)(k0 + 1) * NP_];
        acc = __builtin_amdgcn_wmma_f32_16x16x4_f32(
            /*neg_a=*/false, a, /*neg_b=*/false, b,
            /*c_mod=*/(short)0, acc, /*reuse_a=*/false, /*reuse_b=*/false);
    }

    // epilogue: bias + broadcast to the 16 members of the cluster
    const int np = tn * 16 + l;         // n' = c_p*32 + o
    const int cp = np >> 5;
    const int o  = np & 31;
    const float bias = b1[o];
#pragma unroll
    for (int r = 0; r < 8; ++r) {
        const int m   = r + half * 8;
        const int bid = tm * 16 + m;    // batch index
        const float val = acc[r] + bias;
        float* op = out + (size_t)bid * (N_ * OUT_) + cp * OUT_ + o;
#pragma unroll
        for (int s = 0; s < 16; ++s)
            op[s * (C_ * OUT_)] = val;  // n = cp + 16*s  -> stride 512 floats
    }
}

// ---------------------------------------------------------------------------
extern "C" void kernel_launch(void* const* d_in, const int* in_sizes, int n_in,
                              void* d_out, int out_size, void* d_ws, size_t ws_size,
                              hipStream_t stream) {
    (void)in_sizes; (void)n_in; (void)out_size; (void)ws_size;
    const float* x      = (const float*)d_in[0];
    const float* w_diag = (const float*)d_in[1];
    const float* w_off  = (const float*)d_in[2];
    const float* b1     = (const float*)d_in[3];
    float* out = (float*)d_out;

    float* A = (float*)d_ws;            // 64*512   floats = 128 KB
    float* B = A + BS_ * K_;            // 512*512  floats = 1 MB

    build_A<<<(BS_ * K_ + 255) / 256, 256, 0, stream>>>(x, A);
    build_B<<<(K_ * NP_ + 255) / 256, 256, 0, stream>>>(w_diag, w_off, B);

    dim3 grid(NP_ / 16, BS_ / 16);      // 32 x 4 tiles, one wave32 each
    gemm_wmma<<<grid, 32, 0, stream>>>(A, B, b1, out);
}